// RandomDilationErosion_4896262718042
// MI455X (gfx1250) — compile-verified
//
#include <hip/hip_runtime.h>
#include <cstdint>
#include <cstddef>

// Problem constants (match reference: B=8, H=160, KS=7 -> radius <= 3)
#define B_    8
#define H_    160
#define WPR   5                          // 160 bits / 32 = 5 words per x-row
#define NVOX  (H_ * H_ * H_)             // 4,096,000 voxels per batch
#define WORDS_PER_BATCH (H_ * H_ * WPR)  // 128,000 words per batch
#define TOTAL_WORDS (B_ * WORDS_PER_BATCH)  // 1,024,000 words (4 MB)
#define TOTAL_VOX   (B_ * NVOX)             // 32,768,000 voxels

// Morph tile geometry
#define ZT     8
#define YT     32
#define R_     3                  // max supported ball radius
#define LZ_    (ZT + 2 * R_)      // 14
#define LY_    (YT + 2 * R_)      // 38
#define LROWW  8                  // padded LDS words per row (5 used)
#define TILE_WORDS (LZ_ * LY_ * LROWW)  // 4256 words = 17,024 B LDS

typedef float v4f __attribute__((ext_vector_type(4)));

__device__ __forceinline__ uint32_t ballot32(bool b) {
#if __has_builtin(__builtin_amdgcn_ballot_w32)
    return __builtin_amdgcn_ballot_w32(b);
#else
    return (uint32_t)__ballot(b);
#endif
}

// -------------------------------------------------------------------------
// Pass 1: bit-pack (input != 0), complemented for erode-batches.
// Each wave32 packs 128 consecutive voxels: 4 coalesced strided loads,
// 4 ballots, one 16-byte mask store from lane 0.
// -------------------------------------------------------------------------
__global__ void __launch_bounds__(256)
pack_kernel(const float* __restrict__ in, const float* __restrict__ prob,
            uint32_t* __restrict__ maskIn) {
    const uint32_t gtid = blockIdx.x * blockDim.x + threadIdx.x;
    const uint32_t wave = gtid >> 5;
    const uint32_t lane = gtid & 31u;
    const uint32_t base = wave * 128u;            // 128 voxels per wave
    const uint32_t b    = base / (uint32_t)NVOX;  // NVOX % 128 == 0
    const float p   = prob[b];
    const bool  inv = (p < -0.001f);              // erode: store complement
    uint32_t words[4];
#pragma unroll
    for (int k = 0; k < 4; ++k) {
        const float v   = in[base + 32u * (uint32_t)k + lane];
        const bool  bit = (v != 0.0f) != inv;
        words[k] = ballot32(bit);
    }
    if (lane == 0) {
        uint4 w4;
        w4.x = words[0]; w4.y = words[1]; w4.z = words[2]; w4.w = words[3];
        *(uint4*)(maskIn + (base >> 5)) = w4;     // 16B-aligned b128 store
    }
}

// -------------------------------------------------------------------------
// Pass 2: bit-wise ball dilation on the packed mask.
// Halo rows are staged into pre-zeroed LDS via CDNA5 async global->LDS loads
// (ASYNCcnt, drained with s_wait_asynccnt), then each thread OR-smears its
// row words with 64-bit funnel shifts. copy-op batches use radius 0.
// -------------------------------------------------------------------------
__global__ void __launch_bounds__(256)
morph_kernel(const uint32_t* __restrict__ maskIn, const float* __restrict__ prob,
             const int* __restrict__ dist, uint32_t* __restrict__ maskOut) {
    __shared__ uint32_t tile[TILE_WORDS];
    const int tid = threadIdx.x;
    const int blk = blockIdx.x;
    const int b    = blk / 100;          // 20 z-tiles * 5 y-tiles per batch
    const int r100 = blk - b * 100;
    const int z0   = (r100 / 5) * ZT;
    const int y0   = (r100 % 5) * YT;

    const float p = prob[b];
    int t = dist[b];
    if (t < 0) t = 0;
    if (t > R_) t = R_;
    const bool active = (p > 0.001f) || (p < -0.001f); // dilate or erode(complement)
    const int  teff   = active ? t : 0;                // radius 0 == identity

    // Zero LDS: out-of-bounds halo stays 0 (== SAME zero padding semantics).
    for (int i = tid; i < TILE_WORDS; i += 256) tile[i] = 0u;
    __syncthreads();

    // Stage in-bounds halo rows: CDNA5 async global->LDS b32 loads (per-lane).
    const uint32_t ldsBase = (uint32_t)(uintptr_t)(&tile[0]);
    for (int r = tid; r < LZ_ * LY_; r += 256) {
        const int lz = r / LY_;
        const int ly = r - lz * LY_;
        const int z  = z0 - R_ + lz;
        const int y  = y0 - R_ + ly;
        if ((unsigned)z < (unsigned)H_ && (unsigned)y < (unsigned)H_) {
            const uint32_t* gRow =
                maskIn + (size_t)b * WORDS_PER_BATCH + ((size_t)z * H_ + y) * WPR;
            const uint32_t lRow = ldsBase + (uint32_t)r * (LROWW * 4u);
#pragma unroll
            for (int w = 0; w < WPR; ++w) {
                asm volatile("global_load_async_to_lds_b32 %0, %1, off"
                             :: "v"(lRow + 4u * (uint32_t)w), "v"(gRow + w)
                             : "memory");
            }
        }
    }
    asm volatile("s_wait_asynccnt 0x0" ::: "memory");
    __syncthreads();

    // 256 threads == 8 (z) x 32 (y); each thread produces one full x-row (5 words).
    const int zo = tid >> 5;
    const int yo = tid & 31;
    const int lz = zo + R_;
    const int ly = yo + R_;

    uint32_t out[WPR];
#pragma unroll
    for (int w = 0; w < WPR; ++w) out[w] = 0u;

    const int t2 = teff * teff;
    for (int dz = -teff; dz <= teff; ++dz) {
        for (int dy = -teff; dy <= teff; ++dy) {
            const int rem = t2 - dz * dz - dy * dy;
            if (rem < 0) continue;
            const int e = (rem >= 9) ? 3 : (rem >= 4) ? 2 : (rem >= 1) ? 1 : 0;
            const uint32_t* row = &tile[(((lz + dz) * LY_) + (ly + dy)) * LROWW];
#pragma unroll
            for (int w = 0; w < WPR; ++w) {
                const uint32_t prev = (w > 0)       ? row[w - 1] : 0u;
                const uint32_t cur  = row[w];
                const uint32_t next = (w < WPR - 1) ? row[w + 1] : 0u;
                uint32_t acc = cur;
                const uint64_t cn = (uint64_t)cur  | ((uint64_t)next << 32);
                const uint64_t pc = (uint64_t)prev | ((uint64_t)cur  << 32);
                for (int s = 1; s <= e; ++s) {
                    acc |= (uint32_t)(cn >> s);          // +s (towards -x)
                    acc |= (uint32_t)(pc >> (32 - s));   // -s (towards +x)
                }
                out[w] |= acc;
            }
        }
    }

    uint32_t* oRow = maskOut + (size_t)b * WORDS_PER_BATCH +
                     ((size_t)(z0 + zo) * H_ + (y0 + yo)) * WPR;
#pragma unroll
    for (int w = 0; w < WPR; ++w) oRow[w] = out[w];
}

// -------------------------------------------------------------------------
// Pass 3: out = in * mask_bit (bits flipped for erode-batches).
// 4 voxels/thread: 16B vector load of input (should hit the 192 MB L2,
// warmed by pass 1), 16B NON-TEMPORAL store of output so the write stream
// does not evict the L2-resident input. Lane-0 prefetch -> global_prefetch_b8.
// -------------------------------------------------------------------------
__global__ void __launch_bounds__(256)
apply_kernel(const float* __restrict__ in, const float* __restrict__ prob,
             const uint32_t* __restrict__ maskOut, float* __restrict__ out) {
    const uint32_t gtid = blockIdx.x * blockDim.x + threadIdx.x;
    const uint32_t v0   = gtid * 4u;                  // 4 voxels per thread
    const uint32_t b    = v0 / (uint32_t)NVOX;        // NVOX % 4 == 0
    const float p = prob[b];
    const uint32_t inv = (p < -0.001f) ? 0xFu : 0u;   // flip nibble for erode
    if ((gtid & 31u) == 0u && (v0 + 131072u) < (uint32_t)TOTAL_VOX)
        __builtin_prefetch(in + v0 + 131072u, 0, 3);
    const uint32_t word = maskOut[gtid >> 3];
    const uint32_t nib  = ((word >> ((gtid & 7u) * 4u)) & 0xFu) ^ inv;
    const v4f vin = *(const v4f*)(in + v0);
    v4f vout;
    vout.x = (nib & 1u) ? vin.x : 0.0f;
    vout.y = (nib & 2u) ? vin.y : 0.0f;
    vout.z = (nib & 4u) ? vin.z : 0.0f;
    vout.w = (nib & 8u) ? vin.w : 0.0f;
    __builtin_nontemporal_store(vout, (v4f*)(out + v0));
}

// -------------------------------------------------------------------------
extern "C" void kernel_launch(void* const* d_in, const int* in_sizes, int n_in,
                              void* d_out, int out_size, void* d_ws, size_t ws_size,
                              hipStream_t stream) {
    (void)in_sizes; (void)n_in; (void)out_size; (void)ws_size;
    const float* in   = (const float*)d_in[0];
    const float* prob = (const float*)d_in[1];
    const int*   dist = (const int*)d_in[2];
    float* out = (float*)d_out;

    uint32_t* maskIn  = (uint32_t*)d_ws;            // 4 MB
    uint32_t* maskOut = maskIn + TOTAL_WORDS;       // 4 MB (needs 8 MB ws total)

    pack_kernel <<<TOTAL_VOX / (4 * 256), 256, 0, stream>>>(in, prob, maskIn);
    morph_kernel<<<B_ * 100,              256, 0, stream>>>(maskIn, prob, dist, maskOut);
    apply_kernel<<<TOTAL_VOX / (4 * 256), 256, 0, stream>>>(in, prob, maskOut, out);
}